// TCSA_63943473103336
// MI455X (gfx1250) — compile-verified
//
#include <hip/hip_runtime.h>
#include <math.h>

// ---------------------------------------------------------------------------
// TCSA attention, fused to 3 full passes over x (16,16,64,64,64) fp32.
//   K1: per-(t,n,c) sum/max over HW (256 MB read)
//   K2: single-wave gate kernel: ta & ca MLPs via V_WMMA_F32_16X16X4_F32
//   K3: per-(n,h,w) mean/max over (t,c) of g*x (256 MB read)
//   K4: 3x3 conv + sigmoid -> sa (tiny)
//   K5: out = relu(sa*g*x)  (256 MB read + 256 MB NT write)
// ---------------------------------------------------------------------------

typedef __attribute__((ext_vector_type(2))) float v2f;
typedef __attribute__((ext_vector_type(4))) float v4f;
typedef __attribute__((ext_vector_type(8))) float v8f;

#define WMMA4(A, B, C) \
  __builtin_amdgcn_wmma_f32_16x16x4_f32(false, (A), false, (B), (short)0, (C), false, false)

__device__ __forceinline__ float sigmoidf_(float v) { return 1.0f / (1.0f + __expf(-v)); }

// ---- Kernel 1: HW pool. One block per (t,n,c); 4096 contiguous floats. ----
__global__ __launch_bounds__(256) void tcsa_pool_hw(const float* __restrict__ x,
                                                    float* __restrict__ sumhw,
                                                    float* __restrict__ maxhw) {
  const int tnc = blockIdx.x;
  const v4f* p = (const v4f*)(x + (size_t)tnc * 4096);
  const int tid = threadIdx.x;
  float s = 0.f, m = -INFINITY;
#pragma unroll
  for (int i = 0; i < 4; ++i) {
    v4f v = p[tid + 256 * i];
    s += v.x + v.y + v.z + v.w;
    m = fmaxf(m, fmaxf(fmaxf(v.x, v.y), fmaxf(v.z, v.w)));
  }
#pragma unroll
  for (int off = 16; off > 0; off >>= 1) {
    s += __shfl_down(s, off, 32);
    m = fmaxf(m, __shfl_down(m, off, 32));
  }
  __shared__ float ss[8], sm[8];
  const int wid = tid >> 5;
  if ((tid & 31) == 0) { ss[wid] = s; sm[wid] = m; }
  __syncthreads();
  if (tid == 0) {
    float S = ss[0], M = sm[0];
#pragma unroll
    for (int w = 1; w < 8; ++w) { S += ss[w]; M = fmaxf(M, sm[w]); }
    sumhw[tnc] = S;
    maxhw[tnc] = M;
  }
}

// ---- Kernel 2: gates. ONE wave (32 lanes); all matmuls on the WMMA pipe. ----
__global__ __launch_bounds__(32) void tcsa_gates(const float* __restrict__ sumhw,
                                                 const float* __restrict__ maxhw,
                                                 const float* __restrict__ ta_w1,
                                                 const float* __restrict__ ta_w2,
                                                 const float* __restrict__ ca_w1,
                                                 const float* __restrict__ ca_w2,
                                                 float* __restrict__ g_out) {
  __shared__ float p_avg[256], p_max[256];   // [t*16+n]
  __shared__ float hsum[16];                 // relu(h_avg)+relu(h_max), time MLP hidden
  __shared__ float ta[256];                  // [t*16+n]
  __shared__ float c_avg[1024], c_max[1024]; // [n*64+c]
  __shared__ float hs2[256];                 // [n*16+j], channel MLP hidden (j>=4 are 0)
  __shared__ float ca[1024];                 // [n*64+c]

  const int lane = threadIdx.x;
  const int row = lane & 15;   // M index of A slots, N index of B/C/D slots
  const int hi  = lane >> 4;   // lane half selects K pair (A/B) / M+8 (C/D)

  // ---- pooled time stats over c (mean over C*H*W, max over C*H*W) ----
  for (int idx = lane; idx < 256; idx += 32) {
    float s = 0.f, m = -INFINITY;
    const float* sp = sumhw + idx * 64;
    const float* mp = maxhw + idx * 64;
#pragma unroll 8
    for (int c = 0; c < 64; ++c) { s += sp[c]; m = fmaxf(m, mp[c]); }
    p_avg[idx] = s * (1.0f / 262144.0f);  // /(C*H*W)
    p_max[idx] = m;
  }
  __syncthreads();

  // ---- ta layer 1: h[t] = sum_m p[t,m]*w1[m]  (K=16 -> 4 chained WMMAs each) ----
  v8f ha = {}, hm = {};
#pragma unroll
  for (int kk = 0; kk < 4; ++kk) {
    const int k0 = kk * 4 + 2 * hi;
    v2f a, am, b;
    a.x = p_avg[row * 16 + k0]; a.y = p_avg[row * 16 + k0 + 1];
    am.x = p_max[row * 16 + k0]; am.y = p_max[row * 16 + k0 + 1];
    b.x = ta_w1[k0]; b.y = ta_w1[k0 + 1];   // B[m,j] = w1[m] broadcast over all j
    ha = WMMA4(a, b, ha);
    hm = WMMA4(am, b, hm);
  }
  if (lane == 0) {
#pragma unroll
    for (int r = 0; r < 8; ++r) hsum[r] = fmaxf(ha[r], 0.f) + fmaxf(hm[r], 0.f);
  }
  if (lane == 16) {
#pragma unroll
    for (int r = 0; r < 8; ++r) hsum[8 + r] = fmaxf(ha[r], 0.f) + fmaxf(hm[r], 0.f);
  }
  __syncthreads();

  // ---- ta layer 2: ta[t,n] = sigmoid(hsum[t]*w2[n])  (one WMMA, K padded to 4) ----
  {
    v2f a, b; v8f d = {};
    a.x = (hi == 0) ? hsum[row] : 0.f;  a.y = 0.f;
    b.x = (hi == 0) ? ta_w2[row] : 0.f; b.y = 0.f;
    d = WMMA4(a, b, d);
#pragma unroll
    for (int r = 0; r < 8; ++r) ta[(r + 8 * hi) * 16 + row] = sigmoidf_(d[r]);
  }
  __syncthreads();

  // ---- channel pooled stats of ta*x (gates > 0 so max factors through) ----
  for (int idx = lane; idx < 1024; idx += 32) {
    const int n = idx >> 6, c = idx & 63;
    float s = 0.f, m = -INFINITY;
#pragma unroll
    for (int t = 0; t < 16; ++t) {
      const float tv = ta[t * 16 + n];
      s += tv * sumhw[(t * 16 + n) * 64 + c];
      m = fmaxf(m, tv * maxhw[(t * 16 + n) * 64 + c]);
    }
    c_avg[idx] = s * (1.0f / 65536.0f);  // /(T*H*W)
    c_max[idx] = m;
  }
  __syncthreads();

  // ---- ca layer 1: H[n,j] = sum_c cp[n,c]*w1[j,c]  (K=64 -> 16 chained WMMAs) ----
  v8f Ha = {}, Hm = {};
#pragma unroll
  for (int kk = 0; kk < 16; ++kk) {
    const int k0 = kk * 4 + 2 * hi;
    v2f a, am, b;
    a.x = c_avg[row * 64 + k0]; a.y = c_avg[row * 64 + k0 + 1];
    am.x = c_max[row * 64 + k0]; am.y = c_max[row * 64 + k0 + 1];
    b.x = (row < 4) ? ca_w1[row * 64 + k0] : 0.f;      // B[c,j] = w1[j,c], j<4
    b.y = (row < 4) ? ca_w1[row * 64 + k0 + 1] : 0.f;
    Ha = WMMA4(a, b, Ha);
    Hm = WMMA4(am, b, Hm);
  }
  // relu+sum directly on identical C/D fragments, park in LDS as [n, j]
#pragma unroll
  for (int r = 0; r < 8; ++r)
    hs2[(r + 8 * hi) * 16 + row] = fmaxf(Ha[r], 0.f) + fmaxf(Hm[r], 0.f);
  __syncthreads();

  // ---- ca layer 2: ca[n,c] = sigmoid(sum_j hs2[n,j]*w2[c,j])  (4 WMMAs) ----
#pragma unroll
  for (int gN = 0; gN < 4; ++gN) {
    const int k0 = 2 * hi;
    const int c = gN * 16 + row;
    v2f a, b; v8f d = {};
    a.x = hs2[row * 16 + k0]; a.y = hs2[row * 16 + k0 + 1];  // cols j>=4 are zero
    b.x = ca_w2[c * 4 + k0];  b.y = ca_w2[c * 4 + k0 + 1];
    d = WMMA4(a, b, d);
#pragma unroll
    for (int r = 0; r < 8; ++r) ca[(r + 8 * hi) * 64 + c] = sigmoidf_(d[r]);
  }
  __syncthreads();

  // ---- combined multiplicative gate g[t,n,c] = ta[t,n]*ca[n,c] ----
  for (int idx = lane; idx < 16384; idx += 32) {
    const int t = idx >> 10, nc = idx & 1023, n = nc >> 6;
    g_out[idx] = ta[t * 16 + n] * ca[nc];
  }
}

// ---- Kernel 3: spatial pool: mean/max over (t,c) of g*x per (n,h,w). ----
__global__ __launch_bounds__(64) void tcsa_spatial_pool(const float* __restrict__ x,
                                                        const float* __restrict__ g,
                                                        float* __restrict__ zmean,
                                                        float* __restrict__ zmax) {
  __shared__ float gt[1024];  // g[t,c] for this n
  const int n = blockIdx.y;
  const int tid = threadIdx.x;
  for (int i = tid; i < 1024; i += 64) {
    const int t = i >> 6, c = i & 63;
    gt[i] = g[(t * 16 + n) * 64 + c];
  }
  __syncthreads();
  const int hw = blockIdx.x * 64 + tid;
  float s = 0.f, m = -INFINITY;
  for (int t = 0; t < 16; ++t) {
    const float* xp = x + (size_t)((t * 16 + n) * 64) * 4096 + hw;
    if (t < 15) __builtin_prefetch(xp + (size_t)64 * 4096, 0, 0);
    const float* gp = gt + t * 64;
#pragma unroll 8
    for (int c = 0; c < 64; ++c) {
      const float v = gp[c] * xp[(size_t)c * 4096];
      s += v;
      m = fmaxf(m, v);
    }
  }
  zmean[n * 4096 + hw] = s * (1.0f / 1024.0f);
  zmax[n * 4096 + hw] = m;
}

// ---- Kernel 4: 3x3 cross-correlation over {mean,max} maps + sigmoid. ----
__global__ __launch_bounds__(256) void tcsa_conv_sa(const float* __restrict__ zmean,
                                                    const float* __restrict__ zmax,
                                                    const float* __restrict__ w,
                                                    float* __restrict__ sa) {
  const int n = blockIdx.y;
  const int hw = blockIdx.x * 256 + threadIdx.x;
  const int h = hw >> 6, wd = hw & 63;
  const float* zc0 = zmean + n * 4096;
  const float* zc1 = zmax + n * 4096;
  float acc = 0.f;
#pragma unroll
  for (int kh = 0; kh < 3; ++kh) {
    const int hh = h + kh - 1;
    if (hh < 0 || hh > 63) continue;
#pragma unroll
    for (int kw = 0; kw < 3; ++kw) {
      const int ww = wd + kw - 1;
      if (ww < 0 || ww > 63) continue;
      acc += zc0[hh * 64 + ww] * w[kh * 3 + kw];
      acc += zc1[hh * 64 + ww] * w[9 + kh * 3 + kw];
    }
  }
  sa[n * 4096 + hw] = sigmoidf_(acc);
}

// ---- Kernel 5: out = relu(sa * g * x), streamed with NT hints. ----
__global__ __launch_bounds__(256) void tcsa_apply(const float* __restrict__ x,
                                                  const float* __restrict__ g,
                                                  const float* __restrict__ sa,
                                                  float* __restrict__ out) {
  const long long e4 = (long long)blockIdx.x * 256 + threadIdx.x;
  const long long base = e4 * 4;
  const int tnc = (int)(base >> 12);
  const int hw = (int)(base & 4095);
  const int n = (tnc >> 6) & 15;
  const float gv = g[tnc];
  const v4f xv = __builtin_nontemporal_load((const v4f*)x + e4);           // last read of x
  const v4f sv = *(const v4f*)(sa + ((size_t)n << 12) + hw);
  v4f o;
  o.x = fmaxf(0.f, gv * sv.x * xv.x);
  o.y = fmaxf(0.f, gv * sv.y * xv.y);
  o.z = fmaxf(0.f, gv * sv.z * xv.z);
  o.w = fmaxf(0.f, gv * sv.w * xv.w);
  __builtin_nontemporal_store(o, (v4f*)out + e4);                          // write-only stream
}

extern "C" void kernel_launch(void* const* d_in, const int* in_sizes, int n_in,
                              void* d_out, int out_size, void* d_ws, size_t ws_size,
                              hipStream_t stream) {
  const float* x     = (const float*)d_in[0];
  const float* ta_w1 = (const float*)d_in[1];
  const float* ta_w2 = (const float*)d_in[2];
  const float* ca_w1 = (const float*)d_in[3];
  const float* ca_w2 = (const float*)d_in[4];
  const float* sa_w  = (const float*)d_in[5];
  float* out = (float*)d_out;
  float* ws = (float*)d_ws;

  float* sumhw = ws;            // 16384  [t,n,c]
  float* maxhw = ws + 16384;    // 16384
  float* g     = ws + 32768;    // 16384  ta*ca
  float* zmean = ws + 49152;    // 65536  [n,hw]
  float* zmax  = ws + 114688;   // 65536
  float* sa    = ws + 180224;   // 65536

  tcsa_pool_hw<<<16384, 256, 0, stream>>>(x, sumhw, maxhw);
  tcsa_gates<<<1, 32, 0, stream>>>(sumhw, maxhw, ta_w1, ta_w2, ca_w1, ca_w2, g);
  tcsa_spatial_pool<<<dim3(64, 16), 64, 0, stream>>>(x, g, zmean, zmax);
  tcsa_conv_sa<<<dim3(16, 16), 256, 0, stream>>>(zmean, zmax, sa_w, sa);
  tcsa_apply<<<65536, 256, 0, stream>>>(x, g, sa, out);
}